// PerQueryForceMatchingLoss_6219112644851
// MI455X (gfx1250) — compile-verified
//
#include <hip/hip_runtime.h>
#include <hip/hip_bf16.h>
#include <math.h>

// ---------------------------------------------------------------------------
// PerQueryForceMatchingLoss on MI455X (gfx1250), fp32 WMMA path.
//
// Math: J = scale * sum_s p_s (v_s - o) k_s^T  with o = P V, kbar = P K.
//   ||J1 - J2||^2 = scale^2 * [ (q11 - 2c11 + n1) + (q22 - 2c22 + n2)
//                               - 2 (q12 - c12a - c12b + x12) ]
//   qXY  = p_X^T ((V_X V_Y^T) ⊙ (K_X K_Y^T)) p_Y        (Gram quadratic forms)
//   cXY  = sum_s p_X[s] (v_s · o_Y)(k_s · kbar_Y)        (via o V^T / kbar K^T GEMMs)
//   nX   = |o_X|^2 |kbar_X|^2,  x12 = (o1·o2)(kb1·kb2)
// All contractions are d=64-wide fp32 GEMMs -> V_WMMA_F32_16X16X4_F32.
// ---------------------------------------------------------------------------

#define B_    2
#define H_    8
#define T_    512
#define S_    512
#define SC_   128
#define D_    64
#define TQ    32
#define NTHR  256
#define SCALE_ 0.125f   // 1/sqrt(64)

typedef float v2f __attribute__((ext_vector_type(2)));
typedef float v8f __attribute__((ext_vector_type(8)));

// C[M x 64] (+)= scl * A[M x 64] * op(B)[64 x 64], all operands in LDS.
// A row-major with leading dim lda. BT=false: B[k*ldb + n]; BT=true: B[n*ldb + k].
// Uses V_WMMA_F32_16X16X4_F32; 16x16 output tiles distributed one per wave.
template<int M, bool BT, bool ACC>
__device__ inline void wgemm(float* __restrict__ C, int ldc,
                             const float* __restrict__ A, int lda,
                             const float* __restrict__ B, int ldb,
                             float scl, int tid)
{
  const int wave = tid >> 5;
  const int lane = tid & 31;
  const int half = lane >> 4;       // K-pair select (lanes 16..31 hold K+2,K+3)
  const int l16  = lane & 15;
  const int tiles = (M / 16) * 4;   // 4 tiles along N=64
  for (int tile = wave; tile < tiles; tile += 8) {
    const int mi = tile >> 2;
    const int ni = tile & 3;
    const int am = mi * 16 + l16;   // A row for this lane (16x4 fp32 A layout)
    const int bn = ni * 16 + l16;   // B col / C col for this lane
    v8f acc = {0.f, 0.f, 0.f, 0.f, 0.f, 0.f, 0.f, 0.f};
#pragma unroll
    for (int kk = 0; kk < 64; kk += 4) {
      const int ka = kk + 2 * half;
      v2f av, bv;
      av.x = A[am * lda + ka];
      av.y = A[am * lda + ka + 1];
      if (BT) {
        bv.x = B[bn * ldb + ka];
        bv.y = B[bn * ldb + ka + 1];
      } else {
        bv.x = B[ka * ldb + bn];
        bv.y = B[(ka + 1) * ldb + bn];
      }
      acc = __builtin_amdgcn_wmma_f32_16x16x4_f32(
          /*neg_a=*/false, av, /*neg_b=*/false, bv,
          /*c_mod=*/(short)0, acc, /*reuse_a=*/false, /*reuse_b=*/false);
    }
#pragma unroll
    for (int r = 0; r < 8; ++r) {
      const int cm = mi * 16 + r + 8 * half;   // C/D 32-bit layout
      const float val = acc[r] * scl;
      if (ACC) C[cm * ldc + bn] += val;
      else     C[cm * ldc + bn]  = val;
    }
  }
}

__device__ inline void load_tile(float* __restrict__ dst,
                                 const float* __restrict__ src,
                                 int nfloats, int tid)
{
  const float4* s4 = reinterpret_cast<const float4*>(src);
  float4* d4 = reinterpret_cast<float4*>(dst);
  for (int i = tid; i < (nfloats >> 2); i += NTHR) d4[i] = s4[i];
}

__device__ inline void zero_lds(float* p, int n, int tid)
{
  for (int i = tid; i < n; i += NTHR) p[i] = 0.f;
}

// Row-wise softmax over P[TQ x cols] in LDS; 8 threads per row.
__device__ inline void softmax_rows(float* __restrict__ P, int cols,
                                    float* __restrict__ red, int tid)
{
  const int r = tid >> 3, g = tid & 7;
  float m = -1e30f;
  for (int c = g; c < cols; c += 8) m = fmaxf(m, P[r * cols + c]);
  red[r * 8 + g] = m;
  __syncthreads();
  float rm = red[r * 8];
#pragma unroll
  for (int i = 1; i < 8; ++i) rm = fmaxf(rm, red[r * 8 + i]);
  __syncthreads();
  float s = 0.f;
  for (int c = g; c < cols; c += 8) {
    const float e = __expf(P[r * cols + c] - rm);
    P[r * cols + c] = e;
    s += e;
  }
  red[r * 8 + g] = s;
  __syncthreads();
  float rs = 0.f;
#pragma unroll
  for (int i = 0; i < 8; ++i) rs += red[r * 8 + i];
  const float inv = 1.f / rs;
  for (int c = g; c < cols; c += 8) P[r * cols + c] *= inv;
  __syncthreads();
}

// acc[r] += sum_n X[r*ldx + xoff + n] * Y[r*64 + n]   (n = 0..63)
__device__ inline void rowdot_add(float* __restrict__ acc,
                                  const float* __restrict__ X, int ldx, int xoff,
                                  const float* __restrict__ Y,
                                  float* __restrict__ red, int tid)
{
  const int r = tid >> 3, g = tid & 7;
  float p = 0.f;
  for (int n = g; n < 64; n += 8) p += X[r * ldx + xoff + n] * Y[r * 64 + n];
  red[r * 8 + g] = p;
  __syncthreads();
  if (g == 0) {
    float s = 0.f;
#pragma unroll
    for (int i = 0; i < 8; ++i) s += red[r * 8 + i];
    acc[r] += s;
  }
  __syncthreads();
}

// acc[r] += sum_n X[r*ldx + xoff + n] * U[r*64 + n] * W[r*64 + n]
__device__ inline void rowtriple_add(float* __restrict__ acc,
                                     const float* __restrict__ X, int ldx, int xoff,
                                     const float* __restrict__ U,
                                     const float* __restrict__ W,
                                     float* __restrict__ red, int tid)
{
  const int r = tid >> 3, g = tid & 7;
  float p = 0.f;
  for (int n = g; n < 64; n += 8)
    p += X[r * ldx + xoff + n] * U[r * 64 + n] * W[r * 64 + n];
  red[r * 8 + g] = p;
  __syncthreads();
  if (g == 0) {
    float s = 0.f;
#pragma unroll
    for (int i = 0; i < 8; ++i) s += red[r * 8 + i];
    acc[r] += s;
  }
  __syncthreads();
}

__launch_bounds__(NTHR, 1)
__global__ void pqfm_kernel(const float* __restrict__ q,
                            const float* __restrict__ k,
                            const float* __restrict__ v,
                            const float* __restrict__ kcg,
                            const float* __restrict__ vcg,
                            float* __restrict__ pql)
{
  // ~226 KB static LDS (gfx1250 WGP has 320 KB).
  __shared__ __align__(16) float lQ [TQ * D_];
  __shared__ __align__(16) float lP1[TQ * S_];
  __shared__ __align__(16) float lP2[TQ * SC_];
  __shared__ __align__(16) float lVa[64 * 64];
  __shared__ __align__(16) float lVb[64 * 64];
  __shared__ __align__(16) float lKa[64 * 64];
  __shared__ __align__(16) float lKb[64 * 64];
  __shared__ __align__(16) float lM1[64 * 64];
  __shared__ __align__(16) float lM2[64 * 64];
  __shared__ __align__(16) float lR [TQ * 64];
  __shared__ __align__(16) float lO1[TQ * 64];
  __shared__ __align__(16) float lB1[TQ * 64];
  __shared__ __align__(16) float lO2[TQ * 64];
  __shared__ __align__(16) float lB2[TQ * 64];
  __shared__ float red[TQ * 8];
  __shared__ float aq11[TQ], aq12[TQ], aq22[TQ];
  __shared__ float ac11[TQ], ac12a[TQ], ac12b[TQ], ac22[TQ];

  const int tid = threadIdx.x;
  const int bx  = blockIdx.x;
  const int qt  = bx & 15;          // T/TQ = 16 query tiles
  const int h   = (bx >> 4) & 7;
  const int b   = bx >> 7;

  const float* Qg  = q   + (((size_t)(b * H_ + h)) * T_ + (size_t)qt * TQ) * D_;
  const float* Kg  = k   + ((size_t)(b * H_ + h)) * S_ * D_;
  const float* Vg  = v   + ((size_t)(b * H_ + h)) * S_ * D_;
  const float* K2g = kcg + (size_t)b * SC_ * D_;
  const float* V2g = vcg + (size_t)b * SC_ * D_;

  if (tid < TQ) {
    aq11[tid] = 0.f; aq12[tid] = 0.f; aq22[tid] = 0.f;
    ac11[tid] = 0.f; ac12a[tid] = 0.f; ac12b[tid] = 0.f; ac22[tid] = 0.f;
  }
  load_tile(lQ, Qg, TQ * D_, tid);
  zero_lds(lO1, TQ * 64, tid); zero_lds(lB1, TQ * 64, tid);
  zero_lds(lO2, TQ * 64, tid); zero_lds(lB2, TQ * 64, tid);
  __syncthreads();

  // ---- dense scores + softmax: P1 = softmax(scale * Q K^T) ----
  for (int J = 0; J < 8; ++J) {
    load_tile(lKa, Kg + (size_t)J * 64 * D_, 64 * D_, tid);
    __syncthreads();
    wgemm<TQ, true, false>(lP1 + J * 64, S_, lQ, D_, lKa, D_, SCALE_, tid);
    __syncthreads();
  }
  softmax_rows(lP1, S_, red, tid);

  // ---- cg scores + softmax ----
  for (int J = 0; J < 2; ++J) {
    load_tile(lKa, K2g + (size_t)J * 64 * D_, 64 * D_, tid);
    __syncthreads();
    wgemm<TQ, true, false>(lP2 + J * 64, SC_, lQ, D_, lKa, D_, SCALE_, tid);
    __syncthreads();
  }
  softmax_rows(lP2, SC_, red, tid);

  // ---- o1 = P1 V, kb1 = P1 K ; o2 = P2 V2, kb2 = P2 K2 ----
  for (int J = 0; J < 8; ++J) {
    load_tile(lVa, Vg + (size_t)J * 64 * D_, 64 * D_, tid);
    load_tile(lKa, Kg + (size_t)J * 64 * D_, 64 * D_, tid);
    __syncthreads();
    wgemm<TQ, false, true>(lO1, 64, lP1 + J * 64, S_, lVa, 64, 1.f, tid);
    wgemm<TQ, false, true>(lB1, 64, lP1 + J * 64, S_, lKa, 64, 1.f, tid);
    __syncthreads();
  }
  for (int J = 0; J < 2; ++J) {
    load_tile(lVa, V2g + (size_t)J * 64 * D_, 64 * D_, tid);
    load_tile(lKa, K2g + (size_t)J * 64 * D_, 64 * D_, tid);
    __syncthreads();
    wgemm<TQ, false, true>(lO2, 64, lP2 + J * 64, SC_, lVa, 64, 1.f, tid);
    wgemm<TQ, false, true>(lB2, 64, lP2 + J * 64, SC_, lKa, 64, 1.f, tid);
    __syncthreads();
  }

  // ---- q11 = p1^T (Gv ⊙ Gk) p1, Gram tiles built on the fly ----
  for (int I = 0; I < 8; ++I) {
    load_tile(lVa, Vg + (size_t)I * 64 * D_, 64 * D_, tid);
    load_tile(lKa, Kg + (size_t)I * 64 * D_, 64 * D_, tid);
    zero_lds(lR, TQ * 64, tid);
    __syncthreads();
    for (int J = 0; J < 8; ++J) {
      load_tile(lVb, Vg + (size_t)J * 64 * D_, 64 * D_, tid);
      load_tile(lKb, Kg + (size_t)J * 64 * D_, 64 * D_, tid);
      __syncthreads();
      wgemm<64, true, false>(lM1, 64, lVb, 64, lVa, 64, 1.f, tid); // V_J V_I^T
      wgemm<64, true, false>(lM2, 64, lKb, 64, lKa, 64, 1.f, tid); // K_J K_I^T
      __syncthreads();
      for (int i = tid; i < 4096; i += NTHR) lM1[i] *= lM2[i];     // Hadamard
      __syncthreads();
      wgemm<TQ, false, true>(lR, 64, lP1 + J * 64, S_, lM1, 64, 1.f, tid);
      __syncthreads();
    }
    rowdot_add(aq11, lP1, S_, I * 64, lR, red, tid);
  }

  // ---- q12 = p1^T (Gv12 ⊙ Gk12) p2 ----
  for (int I = 0; I < 8; ++I) {
    load_tile(lVa, Vg + (size_t)I * 64 * D_, 64 * D_, tid);
    load_tile(lKa, Kg + (size_t)I * 64 * D_, 64 * D_, tid);
    zero_lds(lR, TQ * 64, tid);
    __syncthreads();
    for (int Jc = 0; Jc < 2; ++Jc) {
      load_tile(lVb, V2g + (size_t)Jc * 64 * D_, 64 * D_, tid);
      load_tile(lKb, K2g + (size_t)Jc * 64 * D_, 64 * D_, tid);
      __syncthreads();
      wgemm<64, true, false>(lM1, 64, lVb, 64, lVa, 64, 1.f, tid); // [s2, s1]
      wgemm<64, true, false>(lM2, 64, lKb, 64, lKa, 64, 1.f, tid);
      __syncthreads();
      for (int i = tid; i < 4096; i += NTHR) lM1[i] *= lM2[i];
      __syncthreads();
      wgemm<TQ, false, true>(lR, 64, lP2 + Jc * 64, SC_, lM1, 64, 1.f, tid);
      __syncthreads();
    }
    rowdot_add(aq12, lP1, S_, I * 64, lR, red, tid);
  }

  // ---- q22 = p2^T (Gv22 ⊙ Gk22) p2 ----
  for (int I = 0; I < 2; ++I) {
    load_tile(lVa, V2g + (size_t)I * 64 * D_, 64 * D_, tid);
    load_tile(lKa, K2g + (size_t)I * 64 * D_, 64 * D_, tid);
    zero_lds(lR, TQ * 64, tid);
    __syncthreads();
    for (int J = 0; J < 2; ++J) {
      load_tile(lVb, V2g + (size_t)J * 64 * D_, 64 * D_, tid);
      load_tile(lKb, K2g + (size_t)J * 64 * D_, 64 * D_, tid);
      __syncthreads();
      wgemm<64, true, false>(lM1, 64, lVb, 64, lVa, 64, 1.f, tid);
      wgemm<64, true, false>(lM2, 64, lKb, 64, lKa, 64, 1.f, tid);
      __syncthreads();
      for (int i = tid; i < 4096; i += NTHR) lM1[i] *= lM2[i];
      __syncthreads();
      wgemm<TQ, false, true>(lR, 64, lP2 + J * 64, SC_, lM1, 64, 1.f, tid);
      __syncthreads();
    }
    rowdot_add(aq22, lP2, SC_, I * 64, lR, red, tid);
  }

  // ---- c-terms over dense keys: c11, c12a ----
  for (int I = 0; I < 8; ++I) {
    load_tile(lVa, Vg + (size_t)I * 64 * D_, 64 * D_, tid);
    load_tile(lKa, Kg + (size_t)I * 64 * D_, 64 * D_, tid);
    __syncthreads();
    wgemm<TQ, true, false>(lM1, 64, lO1, 64, lVa, 64, 1.f, tid);   // v_s · o1
    wgemm<TQ, true, false>(lM2, 64, lB1, 64, lKa, 64, 1.f, tid);   // k_s · kb1
    __syncthreads();
    rowtriple_add(ac11, lP1, S_, I * 64, lM1, lM2, red, tid);
    wgemm<TQ, true, false>(lM1, 64, lO2, 64, lVa, 64, 1.f, tid);   // v_s · o2
    wgemm<TQ, true, false>(lM2, 64, lB2, 64, lKa, 64, 1.f, tid);   // k_s · kb2
    __syncthreads();
    rowtriple_add(ac12a, lP1, S_, I * 64, lM1, lM2, red, tid);
  }

  // ---- c-terms over cg keys: c22, c12b ----
  for (int Jc = 0; Jc < 2; ++Jc) {
    load_tile(lVa, V2g + (size_t)Jc * 64 * D_, 64 * D_, tid);
    load_tile(lKa, K2g + (size_t)Jc * 64 * D_, 64 * D_, tid);
    __syncthreads();
    wgemm<TQ, true, false>(lM1, 64, lO2, 64, lVa, 64, 1.f, tid);
    wgemm<TQ, true, false>(lM2, 64, lB2, 64, lKa, 64, 1.f, tid);
    __syncthreads();
    rowtriple_add(ac22, lP2, SC_, Jc * 64, lM1, lM2, red, tid);
    wgemm<TQ, true, false>(lM1, 64, lO1, 64, lVa, 64, 1.f, tid);
    wgemm<TQ, true, false>(lM2, 64, lB1, 64, lKa, 64, 1.f, tid);
    __syncthreads();
    rowtriple_add(ac12b, lP2, SC_, Jc * 64, lM1, lM2, red, tid);
  }

  // ---- rank-1 norm terms and final per-query loss ----
  if (tid < TQ) {
    const int r = tid;
    float oo1 = 0.f, bb1 = 0.f, oo2 = 0.f, bb2 = 0.f, o12 = 0.f, b12 = 0.f;
    for (int d2 = 0; d2 < 64; ++d2) {
      const float a0 = lO1[r * 64 + d2], a1 = lB1[r * 64 + d2];
      const float a2 = lO2[r * 64 + d2], a3 = lB2[r * 64 + d2];
      oo1 += a0 * a0; bb1 += a1 * a1;
      oo2 += a2 * a2; bb2 += a3 * a3;
      o12 += a0 * a2; b12 += a1 * a3;
    }
    const float n1  = oo1 * bb1;
    const float n2  = oo2 * bb2;
    const float x12 = o12 * b12;
    const float t1 = aq11[r] - 2.f * ac11[r]  + n1;
    const float t2 = aq22[r] - 2.f * ac22[r]  + n2;
    const float tx = aq12[r] - ac12a[r] - ac12b[r] + x12;
    const float val =
        (SCALE_ * SCALE_) * (t1 + t2 - 2.f * tx) / (float)(D_ * D_);
    pql[((size_t)(b * H_ + h)) * T_ + (size_t)qt * TQ + r] = val;
  }
}

__global__ void pqfm_finalize(const float* __restrict__ pql,
                              float* __restrict__ out, int N)
{
  __shared__ float s1[NTHR];
  __shared__ float s2[NTHR];
  const int tid = threadIdx.x;
  float a = 0.f, sq = 0.f;
  for (int i = tid; i < N; i += NTHR) {
    const float x = pql[i];
    a += x; sq += x * x;
  }
  s1[tid] = a; s2[tid] = sq;
  __syncthreads();
  for (int st = NTHR / 2; st > 0; st >>= 1) {
    if (tid < st) { s1[tid] += s1[tid + st]; s2[tid] += s2[tid + st]; }
    __syncthreads();
  }
  if (tid == 0) {
    const float mean = s1[0] / (float)N;
    float var = (s2[0] - (float)N * mean * mean) / (float)(N - 1);
    var = fmaxf(var, 0.f);
    out[0] = mean;           // loss
    out[1] = sqrtf(var);     // unbiased std
  }
}

extern "C" void kernel_launch(void* const* d_in, const int* in_sizes, int n_in,
                              void* d_out, int out_size, void* d_ws, size_t ws_size,
                              hipStream_t stream)
{
  (void)in_sizes; (void)n_in; (void)out_size; (void)ws_size;
  const float* q   = (const float*)d_in[0];
  const float* k   = (const float*)d_in[1];
  const float* v   = (const float*)d_in[2];
  const float* kcg = (const float*)d_in[3];
  const float* vcg = (const float*)d_in[4];
  float* out = (float*)d_out;
  float* pql = (float*)d_ws;   // 8192 floats = 32 KB scratch

  const int nblocks = B_ * H_ * (T_ / TQ);   // 256 workgroups
  pqfm_kernel<<<nblocks, NTHR, 0, stream>>>(q, k, v, kcg, vcg, pql);
  pqfm_finalize<<<1, NTHR, 0, stream>>>(pql, out, B_ * H_ * T_);
}